// ModelBlock_59382217834541
// MI455X (gfx1250) — compile-verified
//
#include <hip/hip_runtime.h>
#include <stdint.h>
#include <stddef.h>

#define PP 32
#define SS 4
#define HH 256
#define WW 256
#define HW (HH * WW)      /* 65536  = 2^16 */
#define SHW (SS * HW)     /* 262144 = 2^18 */
#define PSHW (PP * SHW)   /* 8388608 */

// ---- CDNA5 async-to-LDS availability (device pass only) ----
#if defined(__HIP_DEVICE_COMPILE__) && defined(__gfx1250__) && \
    __has_builtin(__builtin_amdgcn_global_load_async_to_lds_b32)
#define HAS_ASYNC_LDS 1
#else
#define HAS_ASYNC_LDS 0
#endif

#if HAS_ASYNC_LDS
typedef __attribute__((address_space(1))) int as1_int;
typedef __attribute__((address_space(3))) int as3_int;
#endif

__device__ __forceinline__ void wait_async_zero() {
#if defined(__HIP_DEVICE_COMPILE__) && defined(__gfx1250__)
#if __has_builtin(__builtin_amdgcn_s_wait_asynccnt)
  __builtin_amdgcn_s_wait_asynccnt(0);
  asm volatile("" ::: "memory");
#else
  asm volatile("s_wait_asynccnt 0" ::: "memory");
#endif
#endif
}

// --------------------------------------------------------------------------
// Kernel 1: per-view warp parameters.
//   A = (K_src @ pose[:3,:3]) @ inv(K_tgt)   (3x3)
//   b = K_src @ pose[:3,3]                   (3)
// params layout: s*12 + [A row-major 0..8, b 9..11]
// --------------------------------------------------------------------------
__global__ void setup_params_kernel(const float* __restrict__ pose,  // (S,4,4)
                                    const float* __restrict__ Ksrc,  // (S,3,3)
                                    const float* __restrict__ Ktgt,  // (3,3)
                                    float* __restrict__ params) {
  int s = threadIdx.x;
  if (s >= SS) return;

  float k[9];
#pragma unroll
  for (int i = 0; i < 9; ++i) k[i] = Ktgt[i];

  // inverse via adjugate / det
  float det = k[0] * (k[4] * k[8] - k[5] * k[7])
            - k[1] * (k[3] * k[8] - k[5] * k[6])
            + k[2] * (k[3] * k[7] - k[4] * k[6]);
  float id = 1.0f / det;
  float inv[9];
  inv[0] =  (k[4] * k[8] - k[5] * k[7]) * id;
  inv[1] = -(k[1] * k[8] - k[2] * k[7]) * id;
  inv[2] =  (k[1] * k[5] - k[2] * k[4]) * id;
  inv[3] = -(k[3] * k[8] - k[5] * k[6]) * id;
  inv[4] =  (k[0] * k[8] - k[2] * k[6]) * id;
  inv[5] = -(k[0] * k[5] - k[2] * k[3]) * id;
  inv[6] =  (k[3] * k[7] - k[4] * k[6]) * id;
  inv[7] = -(k[0] * k[7] - k[1] * k[6]) * id;
  inv[8] =  (k[0] * k[4] - k[1] * k[3]) * id;

  const float* Ks = Ksrc + s * 9;
  const float* Po = pose + s * 16;

  // proj[:3,:4] = Ks @ pose[:3,:4]
  float pr[12];
#pragma unroll
  for (int i = 0; i < 3; ++i)
#pragma unroll
    for (int j = 0; j < 4; ++j) {
      float acc = 0.0f;
#pragma unroll
      for (int kk = 0; kk < 3; ++kk) acc += Ks[i * 3 + kk] * Po[kk * 4 + j];
      pr[i * 4 + j] = acc;
    }

  float* o = params + s * 12;
#pragma unroll
  for (int i = 0; i < 3; ++i)
#pragma unroll
    for (int j = 0; j < 3; ++j)
      o[i * 3 + j] = pr[i * 4 + 0] * inv[0 * 3 + j] +
                     pr[i * 4 + 1] * inv[1 * 3 + j] +
                     pr[i * 4 + 2] * inv[2 * 3 + j];
  o[9]  = pr[0 * 4 + 3];
  o[10] = pr[1 * 4 + 3];
  o[11] = pr[2 * 4 + 3];
}

// --------------------------------------------------------------------------
// Kernel 2: per-pixel over-compositing scan along P.
// Writes one float4 {over.x, over.y, over.z, T} per (p, s, h, w) cell.
// Alphas for all 32 planes are staged through LDS with CDNA5 async-to-LDS
// when available (ASYNCcnt path); otherwise plain coalesced loads.
// --------------------------------------------------------------------------
__global__ __launch_bounds__(256) void scan_kernel(
    const float* __restrict__ colors,  // (P,S,H,W,3)
    const float* __restrict__ alphas,  // (P,S,H,W)
    float4* __restrict__ cells)        // (P,S,H,W)
{
  const int tid = threadIdx.x;
  const int pix = blockIdx.x * 256 + tid;  // flat over (S,H,W)

  float al[PP];
#if HAS_ASYNC_LDS
  __shared__ float s_al[PP * 256];  // 32 KB
#pragma unroll
  for (int p = 0; p < PP; ++p) {
    const float* g = alphas + (size_t)p * SHW + pix;
    float* l = &s_al[p * 256 + tid];
    __builtin_amdgcn_global_load_async_to_lds_b32(
        (as1_int*)g, (as3_int*)l, 0, 0);
  }
  wait_async_zero();
#pragma unroll
  for (int p = 0; p < PP; ++p) al[p] = s_al[p * 256 + tid];
#else
#pragma unroll
  for (int p = 0; p < PP; ++p) al[p] = alphas[(size_t)p * SHW + pix];
#endif

  // Net transmittance: T[d] = prod_{j>d} (1 - al[j]); same multiply order
  // as the reference reverse cumprod.
  float T[PP];
  T[PP - 1] = 1.0f;
#pragma unroll
  for (int p = PP - 2; p >= 0; --p) T[p] = T[p + 1] * (1.0f - al[p + 1]);

  float ox = 0.0f, oy = 0.0f, oz = 0.0f;
#pragma unroll
  for (int p = 0; p < PP; ++p) {
    const float* cp = colors + ((size_t)p * SHW + pix) * 3;
    float cx = cp[0], cy = cp[1], cz = cp[2];
    float a = al[p];
    float ca = 1.0f - a;
    ox = ox * ca + cx * a;
    oy = oy * ca + cy * a;
    oz = oz * ca + cz * a;
    cells[(size_t)p * SHW + pix] = make_float4(ox, oy, oz, T[p]);
  }
}

// --------------------------------------------------------------------------
// Kernel 3: projective warp + bilinear sample of the 10-channel stack.
// out layout: (P, S, 10, H, W)
// --------------------------------------------------------------------------
__global__ __launch_bounds__(256) void warp_kernel(
    const float4* __restrict__ cells,     // (P,S,H,W) {over.xyz, T}
    const float* __restrict__ imgs_src,   // (S,H,W,3)
    const float* __restrict__ params,     // (S,12)
    const float* __restrict__ planes,     // (P,)
    float* __restrict__ out)              // (P,S,10,H,W)
{
  const int idx = blockIdx.x * 256 + threadIdx.x;  // < PSHW
  const int p    = idx >> 18;          // / SHW
  const int rem  = idx & (SHW - 1);
  const int s    = rem >> 16;          // / HW
  const int pixr = rem & (HW - 1);
  const int h    = pixr >> 8;
  const int w    = pixr & (WW - 1);

  const float* pa = params + s * 12;
  const float depth = planes[p];
  const float gx = (float)w, gy = (float)h;

  const float ux = depth * (pa[0] * gx + pa[1] * gy + pa[2]) + pa[9];
  const float uy = depth * (pa[3] * gx + pa[4] * gy + pa[5]) + pa[10];
  const float uz = depth * (pa[6] * gx + pa[7] * gy + pa[8]) + pa[11] + 1e-10f;
  const float ix = ux / uz;
  const float iy = uy / uz;

  const float x0f = floorf(ix), y0f = floorf(iy);
  const float x1f = x0f + 1.0f, y1f = y0f + 1.0f;
  const float wx1 = ix - x0f, wx0 = 1.0f - wx1;
  const float wy1 = iy - y0f, wy0 = 1.0f - wy1;

  int pacc = p - 2;
  if (pacc < 0) pacc = 0;
  const float4* basep = cells + (size_t)p * SHW + (size_t)s * HW;
  const float4* basea = cells + (size_t)pacc * SHW + (size_t)s * HW;
  const float4* baseb = cells + (size_t)(PP - 1) * SHW + (size_t)s * HW;
  const float*  bases = imgs_src + (size_t)s * HW * 3;

  float o[10];
#pragma unroll
  for (int c = 0; c < 10; ++c) o[c] = 0.0f;

  auto tap = [&](float xf, float yf, float wgt) {
    bool valid = (xf >= 0.0f) && (xf <= (float)(WW - 1)) &&
                 (yf >= 0.0f) && (yf <= (float)(HH - 1));
    float we = valid ? wgt : 0.0f;
    int xc = (int)fminf(fmaxf(xf, 0.0f), (float)(WW - 1));
    int yc = (int)fminf(fmaxf(yf, 0.0f), (float)(HH - 1));
    int i = yc * WW + xc;
    float  Tv = ((const float*)(basep + i))[3];
    float4 av = basea[i];
    float4 bv = baseb[i];
    const float* sv = bases + (size_t)i * 3;
    o[0] += we * Tv;
    o[1] += we * av.x; o[2] += we * av.y; o[3] += we * av.z;
    o[4] += we * bv.x; o[5] += we * bv.y; o[6] += we * bv.z;
    o[7] += we * sv[0]; o[8] += we * sv[1]; o[9] += we * sv[2];
  };

  tap(x0f, y0f, wx0 * wy0);
  tap(x1f, y0f, wx1 * wy0);
  tap(x0f, y1f, wx0 * wy1);
  tap(x1f, y1f, wx1 * wy1);

  const size_t ob = ((size_t)(p * SS + s) * 10) * HW + (size_t)pixr;
#pragma unroll
  for (int c = 0; c < 10; ++c) out[ob + (size_t)c * HW] = o[c];
}

// --------------------------------------------------------------------------
extern "C" void kernel_launch(void* const* d_in, const int* in_sizes, int n_in,
                              void* d_out, int out_size, void* d_ws,
                              size_t ws_size, hipStream_t stream) {
  const float* colors     = (const float*)d_in[0];  // (P,S,H,W,3)
  const float* alphas     = (const float*)d_in[1];  // (P,S,H,W)
  const float* imgs_src   = (const float*)d_in[2];  // (S,H,W,3)
  const float* mpi_planes = (const float*)d_in[3];  // (P,)
  const float* pose_tgt   = (const float*)d_in[4];  // (S,4,4)
  const float* intrins_s  = (const float*)d_in[5];  // (S,3,3)
  const float* intrins_t  = (const float*)d_in[6];  // (3,3)
  float* out = (float*)d_out;

  // Workspace: [0,256) params, [256, 256 + P*S*H*W*16) scan cells (float4).
  float*  params = (float*)d_ws;
  float4* cells  = (float4*)((char*)d_ws + 256);

  setup_params_kernel<<<1, 32, 0, stream>>>(pose_tgt, intrins_s, intrins_t,
                                            params);
  scan_kernel<<<SHW / 256, 256, 0, stream>>>(colors, alphas, cells);
  warp_kernel<<<PSHW / 256, 256, 0, stream>>>(cells, imgs_src, params,
                                              mpi_planes, out);
}